// MechanismGrabberTopK_28887950033382
// MI455X (gfx1250) — compile-verified
//
#include <hip/hip_runtime.h>
#include <math.h>

// ---------------------------------------------------------------------------
// MechanismGrabberTopK for MI455X (gfx1250), fp32 WMMA (V_WMMA_F32_16X16X4_F32)
// T=2048 tokens, D=512, M=64 experts, top-K=4.
// B-tiles in LDS use k-pair-interleaved layout Bs[k>>1][n][k&1] so each WMMA
// B-fragment {B[k][n], B[k+1][n]} is one aligned ds_load_b64.
// ---------------------------------------------------------------------------

typedef float v2f __attribute__((ext_vector_type(2)));
typedef float v8f __attribute__((ext_vector_type(8)));

__device__ __forceinline__ v8f wmma4(v2f a, v2f b, v8f c) {
  // D = A(16x4 f32) * B(4x16 f32) + C(16x16 f32)
  return __builtin_amdgcn_wmma_f32_16x16x4_f32(
      /*neg_a=*/false, a, /*neg_b=*/false, b,
      /*c_mod=*/(short)0, c, /*reuse_a=*/false, /*reuse_b=*/false);
}

// ======================= Kernel 1: selector layer 1 ========================
// H[2048,1024] = GELU(context[2048,512] @ sel_W1[512,1024] + sel_b1)
__global__ __launch_bounds__(256) void k_sel1(const float* __restrict__ ctx,
                                              const float* __restrict__ W1,
                                              const float* __restrict__ b1,
                                              float* __restrict__ H) {
  __shared__ float As[64 * 16];       // [row][k]
  __shared__ float Bs[8][128][2];     // [k>>1][n][k&1]
  const int tid  = threadIdx.x;
  const int lane = tid & 31;
  const int wave = tid >> 5;
  const int lr = lane & 15, lh = lane >> 4;
  const int m0 = blockIdx.x * 64;
  const int n0 = blockIdx.y * 128;
  const int wm = (wave & 3) * 16;
  const int wn = (wave >> 2) * 64;

  v8f c[4] = {};

  for (int k0 = 0; k0 < 512; k0 += 16) {
    // stage A: 64x16 floats (float4 per thread)
    {
      const int idx = tid * 4;
      const int row = idx >> 4;
      const int kk  = idx & 15;
      *(float4*)(&As[idx]) =
          *(const float4*)(ctx + (m0 + row) * 512 + k0 + kk);
      // stage B: 16x128 floats, pair-interleaved: thread owns k-pair kp and
      // 4 columns n4..n4+3 (two coalesced float4 row loads, two interleaved
      // float4 LDS stores)
      const int kp = tid >> 5;          // 0..7
      const int n4 = (tid & 31) * 4;    // 0..124
      const float* g0 = W1 + (k0 + 2 * kp) * 1024 + n0 + n4;
      const float4 r0 = *(const float4*)(g0);
      const float4 r1 = *(const float4*)(g0 + 1024);
      *(float4*)(&Bs[kp][n4][0])     = make_float4(r0.x, r1.x, r0.y, r1.y);
      *(float4*)(&Bs[kp][n4 + 2][0]) = make_float4(r0.z, r1.z, r0.w, r1.w);
    }
    __syncthreads();
#pragma unroll
    for (int kk = 0; kk < 16; kk += 4) {
      const int kb = kk + lh * 2;       // always even
      v2f a = *(const v2f*)(&As[(wm + lr) * 16 + kb]);
#pragma unroll
      for (int s = 0; s < 4; ++s) {
        const int n = wn + s * 16 + lr;
        v2f b = *(const v2f*)(&Bs[kb >> 1][n][0]);
        c[s] = wmma4(a, b, c[s]);
      }
    }
    __syncthreads();
  }

  // epilogue: +bias, exact GELU
#pragma unroll
  for (int s = 0; s < 4; ++s) {
    const int n  = n0 + wn + s * 16 + lr;
    const float bias = b1[n];
#pragma unroll
    for (int r = 0; r < 8; ++r) {
      const int row = m0 + wm + lh * 8 + r;
      float v = c[s][r] + bias;
      float g = 0.5f * v * (1.0f + erff(v * 0.70710678118654752f));
      H[row * 1024 + n] = g;
    }
  }
}

// ============ Kernel 2: logits + softmax + top-4 + gates + buckets =========
// One block (64 threads) per token.
__global__ __launch_bounds__(64) void k_router(const float* __restrict__ x,
                                               const float* __restrict__ H,
                                               const float* __restrict__ W2,
                                               const float* __restrict__ b2,
                                               const float* __restrict__ Wg,
                                               const float* __restrict__ bg,
                                               int* __restrict__ counts,
                                               int* __restrict__ lists,
                                               float* __restrict__ wts) {
  const int t = blockIdx.x;
  const int m = threadIdx.x;  // 0..63

  __shared__ float sc[64];
  __shared__ float smax;
  __shared__ int   selIdx[4];
  __shared__ float rw[4];
  __shared__ float gacc[4][16];

  // logit[m] = H[t,:] . W2[:,m] + b2[m]    (K = 1024)
  const float* hrow = H + t * 1024;
  float acc = b2[m];
#pragma unroll 4
  for (int k = 0; k < 1024; ++k) acc += hrow[k] * W2[k * 64 + m];
  sc[m] = acc;
  __syncthreads();

  if (m == 0) {
    float mx = sc[0];
    for (int i = 1; i < 64; ++i) mx = fmaxf(mx, sc[i]);
    smax = mx;
  }
  __syncthreads();
  const float ex = expf(acc - smax);
  sc[m] = ex;
  __syncthreads();

  if (m == 0) {
    // top-4 (first-occurrence tie-break, same as lax.top_k), renorm.
    // softmax denominator cancels in the renormalization.
    float selv[4];
    float ssum = 0.0f;
    for (int k = 0; k < 4; ++k) {
      float best = -1.0f;
      int   bi   = 0;
      for (int i = 0; i < 64; ++i)
        if (sc[i] > best) { best = sc[i]; bi = i; }
      selIdx[k] = bi;
      selv[k]   = best;
      sc[bi]    = -1.0f;
      ssum += best;
    }
    float inv = 1.0f / ssum;
    for (int k = 0; k < 4; ++k) rw[k] = selv[k] * inv;
  }
  __syncthreads();

  // gates for the 4 selected experts: 4 groups of 16 lanes
  {
    const int k = m >> 4, j = m & 15;
    const int e = selIdx[k];
    const float* xr = x + t * 512;
    const float* wg = Wg + e * 512;
    float g = 0.0f;
    for (int d = j; d < 512; d += 16) g += xr[d] * wg[d];
    gacc[k][j] = g;
  }
  __syncthreads();

  if (m < 4) {
    const int e = selIdx[m];
    float g = bg[e];
#pragma unroll
    for (int j = 0; j < 16; ++j) g += gacc[m][j];
    const float sig = 1.0f / (1.0f + expf(-g));
    const float w   = rw[m] * sig;
    const int t4 = t * 4 + m;
    wts[t4] = w;
    const int pos = atomicAdd(&counts[e], 1);
    lists[e * 2048 + pos] = t4;
  }
}

// ================= Kernel 3: sparse expert GEMM (bucketed) =================
// Per (expert e, tile of 16 assignments): gather 16 token rows, compute
// 16x512 = X(16x512) @ W_t[e](512x512), add b_t+char, scale by routing
// weight, scatter into partial[t4, :].
__global__ __launch_bounds__(256) void k_expert(const float* __restrict__ x,
                                                const float* __restrict__ Wt,
                                                const float* __restrict__ bt,
                                                const float* __restrict__ ch,
                                                const int* __restrict__ counts,
                                                const int* __restrict__ lists,
                                                const float* __restrict__ wts,
                                                float* __restrict__ partial) {
  const int e    = blockIdx.x;
  const int tile = blockIdx.y;
  const int cnt  = counts[e];
  if (tile * 16 >= cnt) return;

  __shared__ float As[16 * 16];     // gathered X tile [row][k]
  __shared__ float Bs[8][512][2];   // [k>>1][n][k&1]
  __shared__ int   sTok[16];
  __shared__ float sW[16];

  const int tid  = threadIdx.x;
  const int lane = tid & 31;
  const int wave = tid >> 5;        // 0..7 -> n-range [wave*64, +64)
  const int lr = lane & 15, lh = lane >> 4;
  const int wn = wave * 64;

  if (tid < 16) {
    const int a = tile * 16 + tid;
    if (a < cnt) {
      const int t4 = lists[e * 2048 + a];
      sTok[tid] = t4;
      sW[tid]   = wts[t4];
    } else {
      sTok[tid] = -1;
      sW[tid]   = 0.0f;
    }
  }
  __syncthreads();

  v8f c[4] = {};
  const float* We = Wt + e * 262144;  // 512*512

  for (int k0 = 0; k0 < 512; k0 += 16) {
    // stage A: 256 floats, 1 per thread (gathered by token)
    {
      const int row = tid >> 4;
      const int kk  = tid & 15;
      const int t4  = sTok[row];
      As[tid] = (t4 >= 0) ? x[(t4 >> 2) * 512 + k0 + kk] : 0.0f;
      // stage B: 16x512 floats, pair-interleaved; 4 (kp, n4) units per thread
#pragma unroll
      for (int j = 0; j < 4; ++j) {
        const int u  = j * 256 + tid;
        const int kp = u >> 7;          // 0..7
        const int n4 = (u & 127) * 4;   // 0..508
        const float* g0 = We + (k0 + 2 * kp) * 512 + n4;
        const float4 r0 = *(const float4*)(g0);
        const float4 r1 = *(const float4*)(g0 + 512);
        *(float4*)(&Bs[kp][n4][0])     = make_float4(r0.x, r1.x, r0.y, r1.y);
        *(float4*)(&Bs[kp][n4 + 2][0]) = make_float4(r0.z, r1.z, r0.w, r1.w);
      }
    }
    __syncthreads();
#pragma unroll
    for (int kk = 0; kk < 16; kk += 4) {
      const int kb = kk + lh * 2;
      v2f a = *(const v2f*)(&As[lr * 16 + kb]);
#pragma unroll
      for (int s = 0; s < 4; ++s) {
        const int n = wn + s * 16 + lr;
        v2f b = *(const v2f*)(&Bs[kb >> 1][n][0]);
        c[s] = wmma4(a, b, c[s]);
      }
    }
    __syncthreads();
  }

  // epilogue: (acc + b_t + char) * routing_weight -> partial[t4, n]
#pragma unroll
  for (int s = 0; s < 4; ++s) {
    const int n = wn + s * 16 + lr;
    const float bias = bt[e * 512 + n] + ch[e * 512 + n];
#pragma unroll
    for (int r = 0; r < 8; ++r) {
      const int row = lh * 8 + r;
      const int t4  = sTok[row];
      if (t4 >= 0) partial[t4 * 512 + n] = (c[s][r] + bias) * sW[row];
    }
  }
}

// ==================== Kernel 4: integrate (fused concat) ===================
// out[2048,512] = [x | sum_k partial] (2048x1024) @ int_W[1024,512] + int_b
__global__ __launch_bounds__(256) void k_integrate(const float* __restrict__ x,
                                                   const float* __restrict__ partial,
                                                   const float* __restrict__ Wi,
                                                   const float* __restrict__ bi,
                                                   float* __restrict__ out) {
  __shared__ float As[64 * 16];
  __shared__ float Bs[8][128][2];   // [k>>1][n][k&1]
  const int tid  = threadIdx.x;
  const int lane = tid & 31;
  const int wave = tid >> 5;
  const int lr = lane & 15, lh = lane >> 4;
  const int m0 = blockIdx.x * 64;
  const int n0 = blockIdx.y * 128;
  const int wm = (wave & 3) * 16;
  const int wn = (wave >> 2) * 64;

  v8f c[4] = {};

  for (int k0 = 0; k0 < 1024; k0 += 16) {
    // stage A: 64x16; k<512 -> x, k>=512 -> sum of 4 partial slots
    {
      const int idx = tid * 4;
      const int row = idx >> 4;
      const int kk0 = idx & 15;
      const int t   = m0 + row;
#pragma unroll
      for (int j = 0; j < 4; ++j) {
        const int kg = k0 + kk0 + j;
        float v;
        if (kg < 512) {
          v = x[t * 512 + kg];
        } else {
          const int c2 = kg - 512;
          const float* p = partial + (t * 4) * 512 + c2;
          v = p[0] + p[512] + p[1024] + p[1536];
        }
        As[idx + j] = v;
      }
      // stage B: pair-interleaved
      const int kp = tid >> 5;
      const int n4 = (tid & 31) * 4;
      const float* g0 = Wi + (k0 + 2 * kp) * 512 + n0 + n4;
      const float4 r0 = *(const float4*)(g0);
      const float4 r1 = *(const float4*)(g0 + 512);
      *(float4*)(&Bs[kp][n4][0])     = make_float4(r0.x, r1.x, r0.y, r1.y);
      *(float4*)(&Bs[kp][n4 + 2][0]) = make_float4(r0.z, r1.z, r0.w, r1.w);
    }
    __syncthreads();
#pragma unroll
    for (int kk = 0; kk < 16; kk += 4) {
      const int kb = kk + lh * 2;
      v2f a = *(const v2f*)(&As[(wm + lr) * 16 + kb]);
#pragma unroll
      for (int s = 0; s < 4; ++s) {
        const int n = wn + s * 16 + lr;
        v2f b = *(const v2f*)(&Bs[kb >> 1][n][0]);
        c[s] = wmma4(a, b, c[s]);
      }
    }
    __syncthreads();
  }

#pragma unroll
  for (int s = 0; s < 4; ++s) {
    const int n = n0 + wn + s * 16 + lr;
    const float bias = bi[n];
#pragma unroll
    for (int r = 0; r < 8; ++r) {
      const int row = m0 + wm + lh * 8 + r;
      out[row * 512 + n] = c[s][r] + bias;
    }
  }
}

// ================================ launcher =================================
extern "C" void kernel_launch(void* const* d_in, const int* in_sizes, int n_in,
                              void* d_out, int out_size, void* d_ws, size_t ws_size,
                              hipStream_t stream) {
  const float* x     = (const float*)d_in[0];
  const float* ctx   = (const float*)d_in[1];
  const float* W_t   = (const float*)d_in[2];
  const float* b_t   = (const float*)d_in[3];
  const float* chr   = (const float*)d_in[4];
  const float* W_g   = (const float*)d_in[5];
  const float* b_g   = (const float*)d_in[6];
  const float* selW1 = (const float*)d_in[7];
  const float* selb1 = (const float*)d_in[8];
  const float* selW2 = (const float*)d_in[9];
  const float* selb2 = (const float*)d_in[10];
  const float* intW  = (const float*)d_in[11];
  const float* intb  = (const float*)d_in[12];
  float* out = (float*)d_out;

  // workspace layout
  char* ws = (char*)d_ws;
  float* H       = (float*)(ws);                                   //  8 MB: 2048*1024 f32
  float* partial = (float*)(ws + (size_t)8 * 1024 * 1024);         // 16 MB: 8192*512 f32
  float* wts     = (float*)(ws + (size_t)24 * 1024 * 1024);        // 32 KB: 8192 f32
  int*   lists   = (int*)  (ws + (size_t)24 * 1024 * 1024 + 65536);// 512 KB: 64*2048 i32
  int*   counts  = (int*)  (ws + (size_t)24 * 1024 * 1024 + 65536 + 524288); // 64 i32

  hipMemsetAsync(counts, 0, 64 * sizeof(int), stream);

  k_sel1<<<dim3(32, 8), 256, 0, stream>>>(ctx, selW1, selb1, H);
  k_router<<<2048, 64, 0, stream>>>(x, H, selW2, selb2, W_g, b_g,
                                    counts, lists, wts);
  k_expert<<<dim3(64, 128), 256, 0, stream>>>(x, W_t, b_t, chr,
                                              counts, lists, wts, partial);
  k_integrate<<<dim3(32, 4), 256, 0, stream>>>(x, partial, intW, intb, out);
}